// RNN_24129126269609
// MI455X (gfx1250) — compile-verified
//
#include <hip/hip_runtime.h>
#include <math.h>

typedef __attribute__((ext_vector_type(2))) float v2f;
typedef __attribute__((ext_vector_type(8))) float v8f;
typedef __attribute__((ext_vector_type(4))) int   v4i_t;

#define TILE 64
#define KC 64
#define LDSS (KC + 4)   // stride 68 floats -> conflict-free on 64 banks

#if __has_builtin(__builtin_amdgcn_global_load_async_to_lds_b128)
#define USE_ASYNC_LDS 1
#else
#define USE_ASYNC_LDS 0
#endif

// Wait until at most N async (memory->LDS) ops outstanding for this wave.
template <int N>
__device__ __forceinline__ void wait_async()
{
#if USE_ASYNC_LDS
#if __has_builtin(__builtin_amdgcn_s_wait_asynccnt)
    __builtin_amdgcn_s_wait_asynccnt(N);
#else
    if (N == 0) asm volatile("s_wait_asynccnt 0x0" ::: "memory");
    else        asm volatile("s_wait_asynccnt 0x8" ::: "memory");
#endif
#endif
}

// Stage one 64x64 chunk of A and of B into LDS.
// Async path: GLOBAL_LOAD_ASYNC_TO_LDS_B128, 8 instructions/wave (ASYNCcnt += 8).
// Fallback:   load-all-then-store-all through registers so the 8 global_load_b128
//             issue back-to-back before any wait.
__device__ __forceinline__ void stage_chunk(const float* __restrict__ gA,
                                            const float* __restrict__ gB,
                                            float (* __restrict__ As)[LDSS],
                                            float (* __restrict__ Bs)[LDSS],
                                            int tid, int K)
{
#if USE_ASYNC_LDS
#pragma unroll
    for (int i = 0; i < 4; ++i) {
        const int f   = tid + i * 256;   // float4 slot in 64x64 chunk
        const int row = f >> 4;
        const int c4  = (f & 15) << 2;
        __builtin_amdgcn_global_load_async_to_lds_b128(
            (__attribute__((address_space(1))) v4i_t*)(gA + (size_t)row * K + c4),
            (__attribute__((address_space(3))) v4i_t*)&As[row][c4], 0, 0);
        __builtin_amdgcn_global_load_async_to_lds_b128(
            (__attribute__((address_space(1))) v4i_t*)(gB + (size_t)row * K + c4),
            (__attribute__((address_space(3))) v4i_t*)&Bs[row][c4], 0, 0);
    }
#else
    float4 ra[4], rb[4];
#pragma unroll
    for (int i = 0; i < 4; ++i) {
        const int f   = tid + i * 256;
        const int row = f >> 4;
        const int c4  = (f & 15) << 2;
        ra[i] = *(const float4*)(gA + (size_t)row * K + c4);
        rb[i] = *(const float4*)(gB + (size_t)row * K + c4);
    }
#pragma unroll
    for (int i = 0; i < 4; ++i) {
        const int f   = tid + i * 256;
        const int row = f >> 4;
        const int c4  = (f & 15) << 2;
        *(float4*)&As[row][c4] = ra[i];
        *(float4*)&Bs[row][c4] = rb[i];
    }
#endif
}

// Acc = A[M x K] @ Bm[N x K]^T for a 64x64 tile, double-buffered K chunks.
// MODE 0: C = Acc + bias[n];  MODE 1: C = tanh(C + Acc) in place.
template <int MODE>
__global__ __launch_bounds__(256) void wmma_gemm_kernel(
    const float* __restrict__ A,    // [M x K] row-major
    const float* __restrict__ Bm,   // [N x K] row-major
    const float* __restrict__ bias, // [N]    (MODE 0 only)
    float* __restrict__ C,          // [M x N] row-major
    int M, int N, int K)
{
    __shared__ float As[2][TILE][LDSS];
    __shared__ float Bs[2][TILE][LDSS];

    const int m0   = blockIdx.x * TILE;
    const int n0   = blockIdx.y * TILE;
    const int tid  = threadIdx.x;
    const int wave = tid >> 5;
    const int lane = tid & 31;
    const int ln   = lane & 15;
    const int hi   = lane >> 4;

    // 16 subtiles (4x4 of 16x16) over 8 waves: each wave owns 2 N-subtiles.
    const int mi  = wave & 3;
    const int ni0 = wave >> 2;
    const int ni1 = ni0 + 2;

    v8f acc0 = {};
    v8f acc1 = {};

    const int am  = mi * 16 + ln;
    const int bn0 = ni0 * 16 + ln;
    const int bn1 = ni1 * 16 + ln;

    const float* gA = A  + (size_t)m0 * K;
    const float* gB = Bm + (size_t)n0 * K;
    const int nchunks = K / KC;

    stage_chunk(gA, gB, As[0], Bs[0], tid, K);

    for (int c = 0; c < nchunks; ++c) {
        const int  cur  = c & 1;
        const bool more = (c + 1 < nchunks);
        if (more) {
            // Issue next chunk into the other buffer (fully consumed last iter).
            stage_chunk(gA + (size_t)(c + 1) * KC, gB + (size_t)(c + 1) * KC,
                        As[cur ^ 1], Bs[cur ^ 1], tid, K);
            wait_async<8>();   // 8 outstanding = only next chunk in flight
        } else {
            wait_async<0>();
        }
        __syncthreads();       // all waves' data for As/Bs[cur] has landed

#pragma unroll
        for (int k4 = 0; k4 < KC; k4 += 4) {
            const int kk = k4 + 2 * hi;
            const v2f a  = *(const v2f*)&As[cur][am][kk];
            const v2f b0 = *(const v2f*)&Bs[cur][bn0][kk];
            const v2f b1 = *(const v2f*)&Bs[cur][bn1][kk];
            acc0 = __builtin_amdgcn_wmma_f32_16x16x4_f32(
                false, a, false, b0, (short)0, acc0, false, false);
            acc1 = __builtin_amdgcn_wmma_f32_16x16x4_f32(
                false, a, false, b1, (short)0, acc1, false, false);
        }
        __syncthreads();       // done reading buf[cur] before it is refilled
    }

    // C/D layout: element r of acc -> row (mi*16 + r + 8*hi), col (ni*16 + ln)
    const int crow = m0 + mi * 16 + 8 * hi;
    const int c0   = n0 + ni0 * 16 + ln;
    const int c1   = n0 + ni1 * 16 + ln;
#pragma unroll
    for (int r = 0; r < 8; ++r) {
        const size_t i0 = (size_t)(crow + r) * N + c0;
        const size_t i1 = (size_t)(crow + r) * N + c1;
        if (MODE == 0) {
            C[i0] = acc0[r] + bias[c0];
            C[i1] = acc1[r] + bias[c1];
        } else {
            C[i0] = tanhf(C[i0] + acc0[r]);
            C[i1] = tanhf(C[i1] + acc1[r]);
        }
    }
}

// t = 0: h_{-1} == 0, so out[0] = tanh(xproj[0]) elementwise.
__global__ __launch_bounds__(256) void tanh_inplace_kernel(float* __restrict__ p, int n)
{
    const int i = blockIdx.x * blockDim.x + threadIdx.x;
    if (i < n) p[i] = tanhf(p[i]);
}

extern "C" void kernel_launch(void* const* d_in, const int* in_sizes, int n_in,
                              void* d_out, int out_size, void* d_ws, size_t ws_size,
                              hipStream_t stream)
{
    (void)in_sizes; (void)n_in; (void)out_size; (void)d_ws; (void)ws_size;

    const float* x  = (const float*)d_in[0];  // [L, B, D] = [512,128,512]
    const float* Wx = (const float*)d_in[1];  // [H, D]    = [1024,512]
    const float* bx = (const float*)d_in[2];  // [H]
    const float* Wh = (const float*)d_in[3];  // [H, H]    = [1024,1024]
    float* out = (float*)d_out;               // [L, B, H] = [512,128,1024]

    const int L = 512, B = 128, D = 512, H = 1024;
    const int M1 = L * B;                     // x flattens to [65536, 512]

    // 1) xproj for all timesteps straight into out: out[t] = x_t @ Wx^T + bx
    dim3 g1(M1 / TILE, H / TILE);             // 1024 x 16 blocks
    wmma_gemm_kernel<0><<<g1, 256, 0, stream>>>(x, Wx, bx, out, M1, H, D);

    // 2) t = 0: out[0] = tanh(out[0])
    tanh_inplace_kernel<<<(B * H) / 256, 256, 0, stream>>>(out, B * H);

    // 3) t = 1..L-1: out[t] = tanh(out[t] + out[t-1] @ Wh^T)  (Wh stays hot in L2)
    dim3 g2(B / TILE, H / TILE);              // 2 x 16 blocks per step
    for (int t = 1; t < L; ++t) {
        wmma_gemm_kernel<1><<<g2, 256, 0, stream>>>(
            out + (size_t)(t - 1) * B * H,    // h_{t-1}
            Wh, nullptr,
            out + (size_t)t * B * H,          // xproj[t] in, h_t out (in place)
            B, H, H);
    }
}